// GCNN_Dot_Product_3324304687692
// MI455X (gfx1250) — compile-verified
//
#include <hip/hip_runtime.h>
#include <hip/hip_bf16.h>

typedef float v2f __attribute__((ext_vector_type(2)));
typedef float v8f __attribute__((ext_vector_type(8)));

// ---------------------------------------------------------------------------
// Small utility kernels
// ---------------------------------------------------------------------------
__global__ void fill_f32(float* p, float v, long n) {
  long i = (long)blockIdx.x * blockDim.x + threadIdx.x;
  if (i < n) p[i] = v;
}

__global__ void deg_accum(const int* __restrict__ dst, float* __restrict__ deg, int E) {
  long e = (long)blockIdx.x * blockDim.x + threadIdx.x;
  if (e < E) atomicAdd(&deg[dst[e]], 1.0f);
}

__global__ void deg_to_dinv(float* deg, int n) {
  long i = (long)blockIdx.x * blockDim.x + threadIdx.x;
  if (i < n) deg[i] = rsqrtf(deg[i]);
}

// ---------------------------------------------------------------------------
// out[rows x 64] = A[rows x 64] @ W[64 x 64] (+ bias)
// Native-precision fp32 WMMA: V_WMMA_F32_16X16X4_F32.
// One wave -> 16x64 output tile (4 accumulators of v8f), 8 waves/block.
// W staged in 16KB LDS once per block.
//
// A-frag (16x4 f32): lane l holds row (l&15); VGPR0/1 = K + (l>>4)*2 + {0,1}
// B-frag (4x16 f32): VGPR0/1 = rows K+(l>>4)*2+{0,1}, col = l&15
// C/D (16x16 f32):   VGPR r  = row r + (l>>4)*8,      col = l&15
// ---------------------------------------------------------------------------
__global__ __launch_bounds__(256)
void gemm64_wmma(const float* __restrict__ A, const float* __restrict__ W,
                 const float* __restrict__ bias, float* __restrict__ out, int nRows) {
  __shared__ float sW[64 * 64];
  int tid = threadIdx.x;
  for (int i = tid; i < 64 * 64; i += 256) sW[i] = W[i];
  __syncthreads();

  int wave = tid >> 5;
  int lane = tid & 31;
  long tileRow = ((long)blockIdx.x * 8 + wave) * 16;

  int koff = (lane >> 4) << 1;   // 0 or 2
  int mrow = lane & 15;          // 0..15
  long arow = tileRow + mrow;
  if (arow >= nRows) arow = nRows - 1;   // clamp loads; EXEC stays all-1 for WMMA
  const float* aptr = A + arow * 64 + koff;

  v8f acc[4] = {};
  for (int k0 = 0; k0 < 64; k0 += 4) {
    v2f a;
    a.x = aptr[k0];
    a.y = aptr[k0 + 1];
    int kk = k0 + koff;
#pragma unroll
    for (int n = 0; n < 4; ++n) {
      int col = n * 16 + mrow;
      v2f b;
      b.x = sW[kk * 64 + col];
      b.y = sW[(kk + 1) * 64 + col];
      acc[n] = __builtin_amdgcn_wmma_f32_16x16x4_f32(
          /*neg_a=*/false, a, /*neg_b=*/false, b,
          /*c_mod=*/(short)0, acc[n], /*reuse_a=*/false, /*reuse_b=*/false);
    }
  }

  int colBase = lane & 15;
  int rowOff = (lane >> 4) * 8;
#pragma unroll
  for (int r = 0; r < 8; ++r) {
    long row = tileRow + rowOff + r;
    if (row < nRows) {
#pragma unroll
      for (int n = 0; n < 4; ++n) {
        int col = n * 16 + colBase;
        float v = acc[n][r];
        if (bias) v += bias[col];
        out[row * 64 + col] = v;
      }
    }
  }
}

// ---------------------------------------------------------------------------
// Edge scatter: one thread per (edge, channel). Coalesced row reads of h[src],
// row-contiguous f32 atomics into agg[dst].
// ---------------------------------------------------------------------------
__global__ void scatter_edges(const int* __restrict__ src, const int* __restrict__ dst,
                              const float* __restrict__ h, const float* __restrict__ dinv,
                              float* __restrict__ agg, long nWork) {
  long t = (long)blockIdx.x * blockDim.x + threadIdx.x;
  if (t >= nWork) return;
  long e = t >> 6;
  int c = (int)(t & 63);
  int s = src[e];
  int d = dst[e];
  float w = dinv[s] * dinv[d];
  atomicAdd(&agg[(long)d * 64 + c], h[(long)s * 64 + c] * w);
}

// ---------------------------------------------------------------------------
// Per-node fused: self-loop term + bias, LayerNorm (wave32 shfl reductions),
// ReLU, gate by x. One wave per node, 2 channels per lane. Writes g (may alias h).
// ---------------------------------------------------------------------------
__global__ void node_fuse(const float* __restrict__ agg, const float* __restrict__ h,
                          const float* __restrict__ dinv, const float* __restrict__ b,
                          const float* __restrict__ gamma, const float* __restrict__ beta,
                          const float* __restrict__ x, float* __restrict__ g, int nNodes) {
  int wave = threadIdx.x >> 5;
  int lane = threadIdx.x & 31;
  long node = (long)blockIdx.x * (blockDim.x >> 5) + wave;
  if (node >= nNodes) return;

  float di = dinv[node];
  float selfw = di * di;
  int c0 = lane, c1 = lane + 32;
  long base = node * 64;

  float v0 = agg[base + c0] + h[base + c0] * selfw + b[c0];
  float v1 = agg[base + c1] + h[base + c1] * selfw + b[c1];

  float s = v0 + v1;
#pragma unroll
  for (int off = 16; off > 0; off >>= 1) s += __shfl_xor(s, off, 32);
  float mu = s * (1.0f / 64.0f);

  float d0 = v0 - mu, d1 = v1 - mu;
  float q = d0 * d0 + d1 * d1;
#pragma unroll
  for (int off = 16; off > 0; off >>= 1) q += __shfl_xor(q, off, 32);
  float inv = rsqrtf(q * (1.0f / 64.0f) + 1e-5f);

  float y0 = d0 * inv * gamma[c0] + beta[c0];
  float y1 = d1 * inv * gamma[c1] + beta[c1];
  y0 = fmaxf(y0, 0.0f) * x[base + c0];
  y1 = fmaxf(y1, 0.0f) * x[base + c1];
  g[base + c0] = y0;
  g[base + c1] = y1;
}

// ---------------------------------------------------------------------------
extern "C" void kernel_launch(void* const* d_in, const int* in_sizes, int n_in,
                              void* d_out, int out_size, void* d_ws, size_t ws_size,
                              hipStream_t stream) {
  const float* x     = (const float*)d_in[0];
  const int*   eidx  = (const int*)d_in[1];
  const float* W     = (const float*)d_in[2];
  const float* b     = (const float*)d_in[3];
  const float* gamma = (const float*)d_in[4];
  const float* beta  = (const float*)d_in[5];
  const float* fcW   = (const float*)d_in[6];
  const float* fcb   = (const float*)d_in[7];
  float* out = (float*)d_out;

  int N = in_sizes[0] / 64;
  int E = in_sizes[1] / 2;
  const int* src = eidx;
  const int* dst = eidx + E;

  // workspace layout (256B-aligned slabs): dinv[N], h[N*64], agg[N*64]
  char* ws = (char*)d_ws;
  size_t off = 0;
  float* dinv = (float*)(ws + off);
  off += (((size_t)N * 4) + 255) & ~(size_t)255;
  float* h = (float*)(ws + off);
  off += (((size_t)N * 64 * 4) + 255) & ~(size_t)255;
  float* agg = (float*)(ws + off);
  float* g = h;  // safe in-place reuse in node_fuse

  const int TB = 256;

  // 1) degrees (self loop => start at 1) and zero accumulator
  fill_f32<<<(N + TB - 1) / TB, TB, 0, stream>>>(dinv, 1.0f, N);
  fill_f32<<<(int)(((long)N * 64 + TB - 1) / TB), TB, 0, stream>>>(agg, 0.0f, (long)N * 64);
  deg_accum<<<(E + TB - 1) / TB, TB, 0, stream>>>(dst, dinv, E);
  deg_to_dinv<<<(N + TB - 1) / TB, TB, 0, stream>>>(dinv, N);

  // 2) h = x @ W   (bias applied after aggregation)
  int gemmBlocks = (N + 127) / 128;
  gemm64_wmma<<<gemmBlocks, TB, 0, stream>>>(x, W, nullptr, h, N);

  // 3) edge scatter with symmetric normalization
  long nWork = (long)E * 64;
  scatter_edges<<<(int)((nWork + TB - 1) / TB), TB, 0, stream>>>(src, dst, h, dinv, agg, nWork);

  // 4) self-loop + bias + LayerNorm + ReLU + gating
  node_fuse<<<(N + 7) / 8, TB, 0, stream>>>(agg, h, dinv, b, gamma, beta, x, g, N);

  // 5) out = g @ fcW + fcb
  gemm64_wmma<<<gemmBlocks, TB, 0, stream>>>(g, fcW, fcb, out, N);
}